// Model_6012954214541
// MI455X (gfx1250) — compile-verified
//
#include <hip/hip_runtime.h>
#include <hip/hip_bf16.h>
#include <stdint.h>

// ---------------------------------------------------------------------------
// MI455X (gfx1250) implementation.
// GEMMs (the FLOP-dominant stages) run on V_WMMA_F32_16X16X4_F32, fp32 in/out.
// Each wave32 owns a 32x64 C tile = 8 x (16x16) accumulators; per K-step of 4
// it issues 8 WMMAs reusing 2 A-fragments across 4 B-fragments.
// The K-loop is software-pipelined: fragments for step k+4 are loaded before
// the 8 WMMAs of step k execute, so the s_wait for those loads is covered by
// a full WMMA block instead of stalling at loadcnt 0 right after issue.
// TRANSB/MODE are template params (no per-iteration scalar branching).
// A-row prefetch uses global_prefetch_b8 (speculative, OOB-safe).
// All hot matrices (<=38MB) are L2-resident (192MB L2).
// ---------------------------------------------------------------------------

typedef __attribute__((ext_vector_type(2))) float v2f;
typedef __attribute__((ext_vector_type(8))) float v8f;

#define NTOT   6144
#define NDRUG  3072
#define NEDGE  196608
#define GAMMA  0.03125f
#define KW     0.25f
#define NSLOPE 0.2f

// ------------------------------- WMMA GEMM ---------------------------------

__device__ __forceinline__ v8f wmma_f32(v2f a, v2f b, v8f c) {
  // 8 args: (neg_a, A, neg_b, B, c_mod, C, reuse_a, reuse_b)
  return __builtin_amdgcn_wmma_f32_16x16x4_f32(false, a, false, b, (short)0, c,
                                               false, false);
}

template <int MODE>
__device__ __forceinline__ void store_sub(float* __restrict__ C, int ldc,
                                          int rbase, int cbase, v8f acc,
                                          int lane) {
  // C/D 16x16 f32 layout: VGPR r, lanes 0-15 -> M=r, N=lane; lanes 16-31 -> M=r+8.
  int n = lane & 15;
  int kh = lane >> 4;
#pragma unroll
  for (int r = 0; r < 8; ++r) {
    int row = rbase + r + 8 * kh;
    int col = cbase + n;
    float v = acc[r];
    if (MODE == 0)      C[(size_t)row * ldc + col] = v;
    else if (MODE == 1) C[(size_t)row * ldc + col] = 0.5f * v;
    else                C[(size_t)col * ldc + row] += 0.5f * v;  // transposed acc
  }
}

// C[M,Nc] = A[M,K] @ (TRANSB ? B^T : B);  M mult of 32, Nc mult of 64, K mult of 4.
template <int TRANSB, int MODE>
__global__ __launch_bounds__(256)
void wmma_gemm_kernel(const float* __restrict__ A, int lda,
                      const float* __restrict__ B, int ldb,
                      float* __restrict__ C, int ldc,
                      int M, int Nc, int K) {
  int gt = blockIdx.x * blockDim.x + threadIdx.x;
  int wave = gt >> 5;
  int lane = threadIdx.x & 31;
  int tilesN = Nc >> 6;   // 64-wide N tiles
  int tilesM = M >> 5;
  if (wave >= tilesM * tilesN) return;
  int tm = (wave / tilesN) << 5;
  int tn = (wave % tilesN) << 6;

  int ml = lane & 15;  // row (A) / col (B) within 16
  int kh = lane >> 4;  // K-half select

  v8f acc[2][4];
#pragma unroll
  for (int i = 0; i < 2; ++i)
#pragma unroll
    for (int j = 0; j < 4; ++j) acc[i][j] = (v8f){};

  // A 16x4 fragment: lane(ml,kh) holds A[ml][k+2kh], A[ml][k+2kh+1]
  const float* Ar0 = A + (size_t)(tm + ml) * lda + 2 * kh;
  const float* Ar1 = A + (size_t)(tm + 16 + ml) * lda + 2 * kh;
  const float* Bt[4];
  if (TRANSB) {
#pragma unroll
    for (int j = 0; j < 4; ++j)
      Bt[j] = B + (size_t)(tn + 16 * j + ml) * ldb + 2 * kh;
  }

  auto load_frags = [&](int k, v2f& a0, v2f& a1, v2f* b) {
    a0 = *(const v2f*)(Ar0 + k);
    a1 = *(const v2f*)(Ar1 + k);
    if (TRANSB) {
#pragma unroll
      for (int j = 0; j < 4; ++j) b[j] = *(const v2f*)(Bt[j] + k);
    } else {
      const float* Bp0 = B + (size_t)(k + 2 * kh) * ldb + tn + ml;
      const float* Bp1 = Bp0 + ldb;
#pragma unroll
      for (int j = 0; j < 4; ++j) { b[j][0] = Bp0[16 * j]; b[j][1] = Bp1[16 * j]; }
    }
  };

  // software pipeline: loads for step k+4 issued before WMMAs of step k
  v2f a0, a1, b[4];
  load_frags(0, a0, a1, b);
  int k = 0;
  for (; k + 4 < K; k += 4) {
    v2f na0, na1, nb[4];
    load_frags(k + 4, na0, na1, nb);
    __builtin_prefetch(Ar0 + k + 68, 0, 3);   // global_prefetch_b8, one line ahead
    __builtin_prefetch(Ar1 + k + 68, 0, 3);
#pragma unroll
    for (int j = 0; j < 4; ++j) {
      acc[0][j] = wmma_f32(a0, b[j], acc[0][j]);
      acc[1][j] = wmma_f32(a1, b[j], acc[1][j]);
    }
    a0 = na0; a1 = na1;
#pragma unroll
    for (int j = 0; j < 4; ++j) b[j] = nb[j];
  }
#pragma unroll
  for (int j = 0; j < 4; ++j) {   // epilogue: last K-step
    acc[0][j] = wmma_f32(a0, b[j], acc[0][j]);
    acc[1][j] = wmma_f32(a1, b[j], acc[1][j]);
  }

#pragma unroll
  for (int j = 0; j < 4; ++j) {
    store_sub<MODE>(C, ldc, tm,      tn + 16 * j, acc[0][j], lane);
    store_sub<MODE>(C, ldc, tm + 16, tn + 16 * j, acc[1][j], lane);
  }
}

// ----------------------------- small kernels -------------------------------

__device__ __forceinline__ float leaky(float x) { return x > 0.0f ? x : NSLOPE * x; }

__device__ __forceinline__ void edge_sd(const int* __restrict__ ei, int e, int E,
                                        int& s, int& d) {
  if (e < E) { s = ei[e]; d = ei[E + e]; }
  else       { s = e - E; d = e - E; }
}

// ordered-uint encoding for float atomicMax (handles negatives)
__device__ __forceinline__ unsigned enc_key(float f) {
  unsigned u = __float_as_uint(f);
  return (u & 0x80000000u) ? ~u : (u | 0x80000000u);
}
__device__ __forceinline__ float dec_key(unsigned k) {
  unsigned u = (k & 0x80000000u) ? (k & 0x7FFFFFFFu) : ~k;
  return __uint_as_float(u);
}

__global__ void edge_weight_kernel(const int* __restrict__ ei,
                                   const float* __restrict__ adj,
                                   float* __restrict__ w, int E, int N) {
  int e = blockIdx.x * blockDim.x + threadIdx.x;
  if (e >= E + N) return;
  if (e < E) { int s = ei[e]; int d = ei[E + e]; w[e] = adj[(size_t)s * N + d]; }
  else w[e] = 1.0f;
}

__global__ void scores_kernel(const float* __restrict__ XW,
                              const float* __restrict__ as,
                              const float* __restrict__ ad,
                              float* __restrict__ ssrc, float* __restrict__ sdst,
                              int N, int F) {
  int node = (blockIdx.x * blockDim.x + threadIdx.x) >> 5;
  int lane = threadIdx.x & 31;
  if (node >= N) return;
  const float* row = XW + (size_t)node * F;
  float vs = 0.0f, vd = 0.0f;
  for (int f = lane; f < F; f += 32) { float x = row[f]; vs += x * as[f]; vd += x * ad[f]; }
  for (int o = 16; o; o >>= 1) { vs += __shfl_xor(vs, o); vd += __shfl_xor(vd, o); }
  if (lane == 0) { ssrc[node] = vs; sdst[node] = vd; }
}

__global__ void init_node_kernel(unsigned* __restrict__ mu, float* __restrict__ den, int N) {
  int i = blockIdx.x * blockDim.x + threadIdx.x;
  if (i >= N) return;
  mu[i] = enc_key(-__builtin_inff());
  den[i] = 0.0f;
}

__global__ void edge_max_kernel(const int* __restrict__ ei,
                                const float* __restrict__ ssrc,
                                const float* __restrict__ sdst,
                                unsigned* __restrict__ mu, int E, int N) {
  int e = blockIdx.x * blockDim.x + threadIdx.x;
  if (e >= E + N) return;
  int s, d; edge_sd(ei, e, E, s, d);
  float ev = leaky(ssrc[s] + sdst[d]);
  atomicMax(&mu[d], enc_key(ev));
}

__global__ void edge_ex_kernel(const int* __restrict__ ei,
                               const float* __restrict__ ssrc,
                               const float* __restrict__ sdst,
                               const unsigned* __restrict__ mu,
                               const float* __restrict__ w,
                               float* __restrict__ exb, float* __restrict__ den,
                               int E, int N) {
  int e = blockIdx.x * blockDim.x + threadIdx.x;
  if (e >= E + N) return;
  int s, d; edge_sd(ei, e, E, s, d);
  float ev = leaky(ssrc[s] + sdst[d]);
  float ex = __expf(ev - dec_key(mu[d])) * w[e];
  exb[e] = ex;
  atomicAdd(&den[d], ex);
}

__global__ void zero_kernel(float* __restrict__ p, int n) {
  int i = blockIdx.x * blockDim.x + threadIdx.x;
  if (i < n) p[i] = 0.0f;
}

__global__ void copy_kernel(const float* __restrict__ a, float* __restrict__ b, int n) {
  int i = blockIdx.x * blockDim.x + threadIdx.x;
  if (i < n) b[i] = a[i];
}

__global__ void aggregate_kernel(const int* __restrict__ ei,
                                 const float* __restrict__ XW,
                                 const float* __restrict__ exb,
                                 const float* __restrict__ den,
                                 float* __restrict__ outp, int E, int F) {
  int e = blockIdx.x;
  int s, d; edge_sd(ei, e, E, s, d);
  float a = exb[e] / (den[d] + 1e-16f);
  const float* xr = XW + (size_t)s * F;
  float* orow = outp + (size_t)d * F;
  for (int f = threadIdx.x; f < F; f += blockDim.x)
    atomicAdd(&orow[f], a * xr[f]);
}

__global__ void bias_relu_kernel(const float* __restrict__ agg,
                                 const float* __restrict__ bias,
                                 float* __restrict__ H, int n, int F) {
  int i = blockIdx.x * blockDim.x + threadIdx.x;
  if (i >= n) return;
  H[i] = fmaxf(agg[i] + bias[i % F], 0.0f);
}

__global__ void rownorm_kernel(const float* __restrict__ Y, float* __restrict__ Yn,
                               int M, int F) {
  int row = (blockIdx.x * blockDim.x + threadIdx.x) >> 5;
  int lane = threadIdx.x & 31;
  if (row >= M) return;
  const float* y = Y + (size_t)row * F;
  float mn = __builtin_inff(), mx = -__builtin_inff();
  for (int f = lane; f < F; f += 32) { float v = y[f]; mn = fminf(mn, v); mx = fmaxf(mx, v); }
  for (int o = 16; o; o >>= 1) { mn = fminf(mn, __shfl_xor(mn, o)); mx = fmaxf(mx, __shfl_xor(mx, o)); }
  float inv = 1.0f / (mx - mn + 1e-12f);
  float* out = Yn + (size_t)row * F;
  for (int f = lane; f < F; f += 32) out[f] = (y[f] - mn) * inv;
}

__global__ void diag_extract_kernel(const float* __restrict__ K, float* __restrict__ d, int M) {
  int i = blockIdx.x * blockDim.x + threadIdx.x;
  if (i < M) d[i] = K[(size_t)i * M + i];
}

__global__ void meandiag_kernel(const float* __restrict__ d, float* __restrict__ scal, int M) {
  __shared__ float sm[256];
  float s = 0.0f;
  for (int i = threadIdx.x; i < M; i += 256) s += d[i];
  sm[threadIdx.x] = s;
  __syncthreads();
  for (int o = 128; o; o >>= 1) { if (threadIdx.x < o) sm[threadIdx.x] += sm[threadIdx.x + o]; __syncthreads(); }
  if (threadIdx.x == 0) scal[0] = sm[0] / (float)M;
}

__global__ void gip_accum_kernel(float* __restrict__ acc, const float* __restrict__ K,
                                 const float* __restrict__ diag,
                                 const float* __restrict__ scal, int M) {
  int idx = blockIdx.x * blockDim.x + threadIdx.x;
  if (idx >= M * M) return;
  int i = idx / M, j = idx - i * M;
  float invmean = 1.0f / scal[0];
  float dist = (diag[i] + diag[j] - 2.0f * K[idx]) * invmean;
  acc[idx] += __expf(-GAMMA * dist);
}

__global__ void init_minpos_kernel(unsigned* __restrict__ mp) {
  if (threadIdx.x == 0 && blockIdx.x == 0) mp[0] = 0x7F800000u;  // +inf
}

__global__ void absscale_min_kernel(const float* __restrict__ acc, float* __restrict__ B,
                                    unsigned* __restrict__ mp, int n) {
  __shared__ float sm[256];
  int idx = blockIdx.x * blockDim.x + threadIdx.x;
  float v = __builtin_inff();
  if (idx < n) {
    float w = fabsf(KW * acc[idx]);
    B[idx] = w;
    if (w > 0.0f) v = w;
  }
  sm[threadIdx.x] = v;
  __syncthreads();
  for (int o = 128; o; o >>= 1) { if (threadIdx.x < o) sm[threadIdx.x] = fminf(sm[threadIdx.x], sm[threadIdx.x + o]); __syncthreads(); }
  if (threadIdx.x == 0) atomicMin(mp, __float_as_uint(sm[0]));  // non-neg floats: uint order == float order
}

__global__ void zero_replace_kernel(float* __restrict__ B, const unsigned* __restrict__ mp, int n) {
  int i = blockIdx.x * blockDim.x + threadIdx.x;
  if (i >= n) return;
  if (B[i] == 0.0f) B[i] = __uint_as_float(mp[0]);
}

__global__ void diag_sqrt_kernel(const float* __restrict__ B, float* __restrict__ ds, int M) {
  int i = blockIdx.x * blockDim.x + threadIdx.x;
  if (i < M) ds[i] = sqrtf(B[(size_t)i * M + i]);
}

__global__ void norm_div_kernel(float* __restrict__ B, const float* __restrict__ ds, int M) {
  int idx = blockIdx.x * blockDim.x + threadIdx.x;
  if (idx >= M * M) return;
  int i = idx / M, j = idx - i * M;
  B[idx] /= (ds[i] * ds[j]);
}

// ------------------------------- host side ---------------------------------

static inline void launch_gemm(const float* A, int lda, const float* B, int ldb,
                               float* C, int ldc, int M, int Nc, int K,
                               int transB, int mode, hipStream_t s) {
  int tiles = (M / 32) * (Nc / 64);
  int grid = (tiles * 32 + 255) / 256;
  if (transB)
    wmma_gemm_kernel<1, 0><<<grid, 256, 0, s>>>(A, lda, B, ldb, C, ldc, M, Nc, K);
  else if (mode == 0)
    wmma_gemm_kernel<0, 0><<<grid, 256, 0, s>>>(A, lda, B, ldb, C, ldc, M, Nc, K);
  else if (mode == 1)
    wmma_gemm_kernel<0, 1><<<grid, 256, 0, s>>>(A, lda, B, ldb, C, ldc, M, Nc, K);
  else
    wmma_gemm_kernel<0, 2><<<grid, 256, 0, s>>>(A, lda, B, ldb, C, ldc, M, Nc, K);
}

extern "C" void kernel_launch(void* const* d_in, const int* in_sizes, int n_in,
                              void* d_out, int out_size, void* d_ws, size_t ws_size,
                              hipStream_t stream) {
  (void)in_sizes; (void)n_in; (void)out_size; (void)ws_size;
  const float* x       = (const float*)d_in[0];
  const int*   ei      = (const int*)d_in[1];
  const float* adj     = (const float*)d_in[2];
  const float* drugsim = (const float*)d_in[3];
  const float* dissim  = (const float*)d_in[4];
  const float* alpha1  = (const float*)d_in[5];
  const float* alpha2  = (const float*)d_in[6];
  const float* W[3]  = {(const float*)d_in[7],  (const float*)d_in[11], (const float*)d_in[15]};
  const float* AS[3] = {(const float*)d_in[8],  (const float*)d_in[12], (const float*)d_in[16]};
  const float* AD[3] = {(const float*)d_in[9],  (const float*)d_in[13], (const float*)d_in[17]};
  const float* BI[3] = {(const float*)d_in[10], (const float*)d_in[14], (const float*)d_in[18]};
  const int Fdim[3] = {512, 256, 128};
  float* out = (float*)d_out;

  const int N = NTOT, E = NEDGE, EA = NEDGE + NTOT, M = NDRUG;

  // workspace layout
  float* ws = (float*)d_ws;
  size_t off = 0;
  auto alloc = [&](size_t n) { float* p = ws + off; off += n; return p; };
  float*    XW   = alloc((size_t)N * 512);
  float*    AGG  = alloc((size_t)N * 512);
  float*    H[3] = {alloc((size_t)N * 512), alloc((size_t)N * 256), alloc((size_t)N * 128)};
  float*    WE   = alloc(EA);
  float*    EX   = alloc(EA);
  float*    SSRC = alloc(N);
  float*    SDST = alloc(N);
  unsigned* MU   = (unsigned*)alloc(N);
  float*    DEN  = alloc(N);
  float*    YN   = alloc((size_t)M * 512);
  float*    KM   = alloc((size_t)M * M);
  float*    ACCD = alloc((size_t)M * M);
  float*    ACCS = alloc((size_t)M * M);
  float*    DIAG = alloc(M);
  float*    DS   = alloc(M);
  float*    SCAL = alloc(16);
  unsigned* MINP = (unsigned*)(SCAL) + 1;

  // edge weights (shared by all layers)
  edge_weight_kernel<<<(EA + 255) / 256, 256, 0, stream>>>(ei, adj, WE, E, N);

  // -------- 3 GAT layers --------
  const float* Xin = x;
  int Fin = N;
  for (int l = 0; l < 3; ++l) {
    int F = Fdim[l];
    launch_gemm(Xin, Fin, W[l], F, XW, F, N, F, Fin, 0, 0, stream);
    scores_kernel<<<(N * 32) / 256, 256, 0, stream>>>(XW, AS[l], AD[l], SSRC, SDST, N, F);
    init_node_kernel<<<(N + 255) / 256, 256, 0, stream>>>(MU, DEN, N);
    edge_max_kernel<<<(EA + 255) / 256, 256, 0, stream>>>(ei, SSRC, SDST, MU, E, N);
    edge_ex_kernel<<<(EA + 255) / 256, 256, 0, stream>>>(ei, SSRC, SDST, MU, WE, EX, DEN, E, N);
    zero_kernel<<<((N * F) + 255) / 256, 256, 0, stream>>>(AGG, N * F);
    aggregate_kernel<<<EA, (F < 256 ? F : 256), 0, stream>>>(ei, XW, EX, DEN, AGG, E, F);
    bias_relu_kernel<<<((N * F) + 255) / 256, 256, 0, stream>>>(AGG, BI[l], H[l], N * F, F);
    Xin = H[l];
    Fin = F;
  }

  // -------- GIP kernels, accumulated per half --------
  copy_kernel<<<((M * M) + 255) / 256, 256, 0, stream>>>(drugsim, ACCD, M * M);
  copy_kernel<<<((M * M) + 255) / 256, 256, 0, stream>>>(dissim,  ACCS, M * M);

  for (int l = 0; l < 3; ++l) {
    int F = Fdim[l];
    for (int half = 0; half < 2; ++half) {
      const float* Y = H[l] + (size_t)(half ? NDRUG : 0) * F;
      float* acc = half ? ACCS : ACCD;
      rownorm_kernel<<<(M * 32) / 256, 256, 0, stream>>>(Y, YN, M, F);
      launch_gemm(YN, F, YN, F, KM, M, M, M, F, /*transB=*/1, 0, stream);
      diag_extract_kernel<<<(M + 255) / 256, 256, 0, stream>>>(KM, DIAG, M);
      meandiag_kernel<<<1, 256, 0, stream>>>(DIAG, SCAL, M);
      gip_accum_kernel<<<((M * M) + 255) / 256, 256, 0, stream>>>(acc, KM, DIAG, SCAL, M);
    }
  }

  // -------- normalized kernels + final GEMMs --------
  for (int half = 0; half < 2; ++half) {
    const float* acc = half ? ACCS : ACCD;
    const float* alpha = half ? alpha2 : alpha1;
    int mode = half ? 2 : 1;  // drug half first (plain 0.5x store), dis half adds 0.5*out2^T
    init_minpos_kernel<<<1, 32, 0, stream>>>(MINP);
    absscale_min_kernel<<<((M * M) + 255) / 256, 256, 0, stream>>>(acc, KM, MINP, M * M);
    zero_replace_kernel<<<((M * M) + 255) / 256, 256, 0, stream>>>(KM, MINP, M * M);
    diag_sqrt_kernel<<<(M + 255) / 256, 256, 0, stream>>>(KM, DS, M);
    norm_div_kernel<<<((M * M) + 255) / 256, 256, 0, stream>>>(KM, DS, M);
    launch_gemm(KM, M, alpha, M, out, M, M, M, M, 0, mode, stream);
  }
}